// TimeMHA_2018634629877
// MI455X (gfx1250) — compile-verified
//
#include <hip/hip_runtime.h>

#define B_ 2
#define T_ 2048
#define D_ 256
#define H_ 8
#define KC 256          // K-chunk staged in LDS
#define KCP 258         // padded row pitch (stride-2 banks -> conflict free; 8B aligned)

typedef __attribute__((ext_vector_type(2))) float v2f;
typedef __attribute__((ext_vector_type(8))) float v8f;
typedef int v4i_ __attribute__((vector_size(16)));                 // matches builtin param
typedef __attribute__((address_space(1))) v4i_* gas_v4i;           // __device__ (global)
typedef __attribute__((address_space(3))) v4i_* las_v4i;           // __shared__ (LDS)

#if defined(__has_builtin)
#  if __has_builtin(__builtin_amdgcn_global_load_async_to_lds_b128)
#    define HAVE_ASYNC_LDS 1
#  endif
#endif

__device__ __forceinline__ float lrelu(float x) { return x > 0.f ? x : 0.01f * x; }

// ---------------- Kernel 1: per-head scalar projections q,k : [B,H,T] ----------------
__global__ void proj_qk(const float* __restrict__ query, const float* __restrict__ key,
                        const float* __restrict__ Wq, const float* __restrict__ bq,
                        const float* __restrict__ Wk, const float* __restrict__ bk,
                        float* __restrict__ qws, float* __restrict__ kws) {
  int gid = blockIdx.x * blockDim.x + threadIdx.x;   // [0, B*H*T), layout (b,h,t)
  int t  = gid % T_;
  int bh = gid / T_;
  int h  = bh % H_;
  int b  = bh / H_;
  const float4* qr = (const float4*)(query + (size_t)(b * T_ + t) * D_);
  const float4* kr = (const float4*)(key   + (size_t)(b * T_ + t) * D_);
  const float4* wq = (const float4*)(Wq + h * D_);
  const float4* wk = (const float4*)(Wk + h * D_);
  float qa = bq[h], ka = bk[h];
#pragma unroll 8
  for (int i = 0; i < D_ / 4; ++i) {
    float4 a = qr[i], w = wq[i];
    qa += a.x * w.x + a.y * w.y + a.z * w.z + a.w * w.w;
    float4 c = kr[i], u = wk[i];
    ka += c.x * u.x + c.y * u.y + c.z * u.z + c.w * u.w;
  }
  qws[gid] = qa;
  kws[gid] = ka;
}

// ---------------- Kernel 2a: causal prefix max of k per (b,h) ----------------
__global__ void kprefix(const float* __restrict__ kws, float* __restrict__ kmax) {
  int bh = threadIdx.x;                 // 16 threads total
  const float* kr = kws + (size_t)bh * T_;
  float*       mr = kmax + (size_t)bh * T_;
  float m = -3.4e38f;
  for (int t = 0; t < T_; ++t) { m = fmaxf(m, kr[t]); mr[t] = m; }
}

// ---------------- Kernel 2b: per-row softmax max + 1/denominator ----------------
__global__ void rowstats(const float* __restrict__ qws, const float* __restrict__ kws,
                         const float* __restrict__ kmax,
                         float* __restrict__ mrow, float* __restrict__ rinv) {
  int wid  = (blockIdx.x * blockDim.x + threadIdx.x) >> 5;  // one wave32 per row
  int lane = threadIdx.x & 31;
  int t  = wid % T_;
  int bh = wid / T_;
  const float* kr = kws + (size_t)bh * T_;
  float q = qws[(size_t)bh * T_ + t];
  float m = lrelu(q + kmax[(size_t)bh * T_ + t]);   // lrelu monotonic -> exact row max
  float s = 0.f;
  for (int j = lane; j <= t; j += 32)               // trip count uniform per wave
    s += __expf(lrelu(q + kr[j]) - m);
  for (int off = 16; off; off >>= 1)
    s += __shfl_xor(s, off, 32);
  if (lane == 0) { mrow[wid] = m; rinv[wid] = 1.f / s; }
}

// ---------------- Kernel 3: fused a-tile generation + NT store + AV WMMA ----------------
// block = 512 threads (16 wave32), grid = B*H*(T/16). Each block: one (b,h) 16-row tile.
__launch_bounds__(512)
__global__ void attn_av(const float* __restrict__ value,
                        const float* __restrict__ qws, const float* __restrict__ kws,
                        const float* __restrict__ mrow, const float* __restrict__ rinv,
                        float* __restrict__ out) {
  __shared__ __align__(16) float aT[16 * KCP];
  __shared__ __align__(16) float kT[KC];
  __shared__ float qL[16], mL[16], iL[16];

  int tid  = threadIdx.x;
  int lane = tid & 31;
  int wv   = tid >> 5;                 // 0..15 -> output column group
  int tt   = blockIdx.x & 127;
  int bh   = blockIdx.x >> 7;
  int h    = bh & (H_ - 1);
  int b    = bh >> 3;
  int t0   = tt * 16;
  size_t bhT = (size_t)bh * T_;

  if (tid < 16) {
    qL[tid] = qws [bhT + t0 + tid];
    mL[tid] = mrow[bhT + t0 + tid];
    iL[tid] = rinv[bhT + t0 + tid];
  }
  __syncthreads();

  const float* kr = kws + bhT;
  const size_t OUT_BASE = (size_t)B_ * T_ * H_ * T_;   // start of 'outputs' region

  v8f acc = {0.f, 0.f, 0.f, 0.f, 0.f, 0.f, 0.f, 0.f};
  int n0   = wv * 16;
  int mA   = lane & 15;                 // A-matrix row (and B-matrix column) per ISA layout
  int ko   = (lane < 16) ? 0 : 2;       // K sub-offset per lane half
  int tmax = t0 + 15;

  for (int c = 0; c < T_ / KC; ++c) {
    int s0 = c * KC;
    bool active = (s0 <= tmax);         // block-uniform

    if (active) {
      // ---- stage k chunk (1 KB) into LDS via async global->LDS DMA ----
      if (tid < KC / 4) {
#ifdef HAVE_ASYNC_LDS
        __builtin_amdgcn_global_load_async_to_lds_b128(
            (gas_v4i)(kr + s0 + 4 * tid), (las_v4i)&kT[4 * tid], 0, 0);
#else
        *(float4*)&kT[4 * tid] = *(const float4*)(kr + s0 + 4 * tid);
#endif
      }
#ifdef HAVE_ASYNC_LDS
#  if __has_builtin(__builtin_amdgcn_s_wait_asynccnt)
      __builtin_amdgcn_s_wait_asynccnt(0);
#  else
      asm volatile("s_wait_asynccnt 0" ::: "memory");
#  endif
#endif
      __syncthreads();

      // ---- phase 1: branchless a tile; NT store to d_out + LDS store ----
#pragma unroll
      for (int e = 0; e < 16 * KC; e += 512) {
        int idx = e + tid;
        int r = idx >> 8;               // tile row (uniform per wave)
        int s = idx & (KC - 1);
        int t  = t0 + r;
        int sg = s0 + s;
        float p   = __expf(lrelu(qL[r] + kT[s]) - mL[r]) * iL[r];
        float val = (sg <= t) ? p : 0.f;   // v_cndmask, no EXEC churn
        __builtin_nontemporal_store(val, &out[(((size_t)(b * T_ + t) * H_) + h) * T_ + sg]);
        aT[r * KCP + s] = val;
      }
      __syncthreads();

      // ---- phase 2: each wave accumulates 16x16 C with f32 WMMA over this K-chunk ----
      const float* vbase = value + ((size_t)b * T_ + s0) * D_ + n0 + mA;
#pragma unroll 4
      for (int kk = 0; kk < KC; kk += 4) {
        v2f av = *(const v2f*)&aT[mA * KCP + kk + ko];   // A: 16x4, 2 f32 per lane
        const float* bp = vbase + (size_t)(kk + ko) * D_;
        v2f bv;
        bv.x = bp[0];                                    // B: 4x16, rows kk+ko, kk+ko+1
        bv.y = bp[D_];
        acc = __builtin_amdgcn_wmma_f32_16x16x4_f32(
            /*neg_a=*/false, av, /*neg_b=*/false, bv,
            /*c_mod=*/(short)0, acc, /*reuse_a=*/false, /*reuse_b=*/false);
      }
      __syncthreads();   // protect aT/kT before next chunk overwrites them
    } else {
      // ---- fully-masked chunk: stream zeros only ----
#pragma unroll
      for (int e = 0; e < 16 * KC; e += 512) {
        int idx = e + tid;
        int r = idx >> 8;
        int s = idx & (KC - 1);
        int t  = t0 + r;
        int sg = s0 + s;
        __builtin_nontemporal_store(0.f, &out[(((size_t)(b * T_ + t) * H_) + h) * T_ + sg]);
      }
    }
  }

  // ---- write C (16x16 f32 layout: VGPR r -> M = r + (lane<16 ? 0 : 8), N = lane&15) ----
  int nC   = lane & 15;
  int mOff = (lane < 16) ? 0 : 8;
  size_t obase = OUT_BASE + (size_t)(b * T_ + t0 + mOff) * (H_ * D_) + (size_t)h * D_ + n0 + nC;
#pragma unroll
  for (int r = 0; r < 8; ++r)
    __builtin_nontemporal_store(acc[r], &out[obase + (size_t)r * (H_ * D_)]);
}

extern "C" void kernel_launch(void* const* d_in, const int* in_sizes, int n_in,
                              void* d_out, int out_size, void* d_ws, size_t ws_size,
                              hipStream_t stream) {
  const float* query = (const float*)d_in[0];
  const float* key   = (const float*)d_in[1];
  const float* value = (const float*)d_in[2];
  const float* Wq    = (const float*)d_in[3];
  const float* bq    = (const float*)d_in[4];
  const float* Wk    = (const float*)d_in[5];
  const float* bk    = (const float*)d_in[6];
  float* out = (float*)d_out;

  float* ws   = (float*)d_ws;          // 5 * 32768 floats = 640 KB scratch
  float* qws  = ws;
  float* kws  = ws + 32768;
  float* kmax = ws + 65536;
  float* mrow = ws + 98304;
  float* rinv = ws + 131072;

  proj_qk <<<B_ * H_ * T_ / 256, 256, 0, stream>>>(query, key, Wq, bq, Wk, bk, qws, kws);
  kprefix <<<1, B_ * H_, 0, stream>>>(kws, kmax);
  rowstats<<<B_ * H_ * T_ / 8, 256, 0, stream>>>(qws, kws, kmax, mrow, rinv);
  attn_av <<<B_ * H_ * (T_ / 16), 512, 0, stream>>>(value, qws, kws, mrow, rinv, out);
}